// AdamOptimizer_3427383902676
// MI455X (gfx1250) — compile-verified
//
#include <hip/hip_runtime.h>
#include <cmath>

// Sparse Adam (t=1) on MI455X / gfx1250.
// d_in: param(500000x128 f32), m, v, grad_values(262144x128 f32), grad_indices(262144 i64)
// d_out: [param_new | m_new | v_new], each n_rows*128 f32.
//
// Strategy: inverse map row->grad slot in d_ws, then ONE streaming pass over all
// rows (wave32 = one 512B row, lane = one float4). Rows are double-buffered
// through LDS with gfx1250 async global->LDS loads (ASYNCcnt), giving each wave
// 2 rows (3KB) in flight without VGPR pressure; outputs use non-temporal b128
// stores (every byte touched exactly once -> don't thrash the 192MB L2).
// ~1.67 GB total traffic -> ~72us at 23.3 TB/s.

typedef float v4f __attribute__((ext_vector_type(4)));
typedef int v4i __attribute__((vector_size(4 * sizeof(int))));  // matches builtin param type

#define ADAM_DIM 128
#define WPB 8  // waves per 256-thread block (wave32)

#if defined(__has_builtin)
#if __has_builtin(__builtin_amdgcn_global_load_async_to_lds_b128)
#define HAVE_GFX1250_ASYNC 1
#endif
#endif

#ifdef HAVE_GFX1250_ASYNC
#define AS1 __attribute__((address_space(1)))
#define AS3 __attribute__((address_space(3)))
// Builtin signature (from hipcc diagnostic): param0 = v4i addrspace(1)* (global src),
// param1 = v4i addrspace(3)* (LDS dst), then imm offset, imm cpol.
#define ASYNC_B128(gsrc, ldst) \
  __builtin_amdgcn_global_load_async_to_lds_b128((AS1 v4i*)(gsrc), (AS3 v4i*)(ldst), 0, 0)
#if __has_builtin(__builtin_amdgcn_s_wait_asynccnt)
#define WAIT_ASYNC(n) __builtin_amdgcn_s_wait_asynccnt(n)
#else
#define WAIT_ASYNC(n) asm volatile("s_wait_asynccnt %0" ::"n"(n) : "memory")
#endif
#endif

// ---------------------------------------------------------------- map build
__global__ __launch_bounds__(256) void init_map_kernel(int* __restrict__ map, int n) {
  int i = blockIdx.x * 256 + threadIdx.x;
  if (i < n) map[i] = -1;
}

__global__ __launch_bounds__(256) void build_map_kernel(const long long* __restrict__ idx,
                                                        int* __restrict__ map, int nnz) {
  int i = blockIdx.x * 256 + threadIdx.x;
  if (i < nnz) map[(int)idx[i]] = i;
}

// ---------------------------------------------------------------- Adam math
__device__ __forceinline__ void adam_update4(v4f& p, v4f& mm, v4f& vv, v4f g, float lr_t) {
  const float one_m_b1 = 0.1f;    // f32(1 - 0.9)
  const float one_m_b2 = 0.001f;  // f32(1 - 0.999)
  const float bc1 = 0.1f;         // 1 - b1^1
  const float bc2 = 0.001f;       // 1 - b2^1
#pragma unroll
  for (int c = 0; c < 4; ++c) {
    float mo = mm[c] + one_m_b1 * (g[c] - mm[c]);
    float vo = vv[c] + one_m_b2 * (g[c] * g[c] - vv[c]);
    float mt = mo / bc1;
    float vt = vo / bc2;
    p[c] = p[c] - lr_t * mt / (sqrtf(vt) + 1e-8f);
    mm[c] = mo;
    vv[c] = vo;
  }
}

// ---------------------------------------------------------------- fused pass
#ifdef HAVE_GFX1250_ASYNC
// Async global->LDS double-buffered streaming pass.
__global__ __launch_bounds__(256) void adam_rows_kernel(
    const float* __restrict__ p_in, const float* __restrict__ m_in,
    const float* __restrict__ v_in, const float* __restrict__ g_in,
    const int* __restrict__ map,
    float* __restrict__ p_out, float* __restrict__ m_out, float* __restrict__ v_out,
    int n_rows, float lr_t) {
  __shared__ v4f lp[2][WPB][32];
  __shared__ v4f lm[2][WPB][32];
  __shared__ v4f lv[2][WPB][32];

  const int lane = threadIdx.x & 31;
  const int w = threadIdx.x >> 5;
  const int wave0 = blockIdx.x * WPB + w;
  const int stride = gridDim.x * WPB;
  if (wave0 >= n_rows) return;  // wave-uniform, per-wave LDS only, no barriers used

  // Prologue: stage first row into buffer 0 (per-lane 16B -> 512B/row/array).
  {
    const size_t b0 = (size_t)wave0 * ADAM_DIM + (size_t)lane * 4;
    ASYNC_B128(p_in + b0, &lp[0][w][lane]);
    ASYNC_B128(m_in + b0, &lm[0][w][lane]);
    ASYNC_B128(v_in + b0, &lv[0][w][lane]);
  }

  int buf = 0;
  for (int row = wave0; row < n_rows; row += stride, buf ^= 1) {
    const int nrow = row + stride;
    if (nrow < n_rows) {
      // Stage next row into the other buffer, then wait for the current row's
      // 3 loads (async loads complete in order -> asynccnt<=3 suffices).
      const size_t nb = (size_t)nrow * ADAM_DIM + (size_t)lane * 4;
      ASYNC_B128(p_in + nb, &lp[buf ^ 1][w][lane]);
      ASYNC_B128(m_in + nb, &lm[buf ^ 1][w][lane]);
      ASYNC_B128(v_in + nb, &lv[buf ^ 1][w][lane]);
      // Depth-2 prefetch: start HBM lines for the row after next (global_prefetch_b8).
      const int ppro = nrow + stride;
      if (ppro < n_rows) {
        const size_t pb = (size_t)ppro * ADAM_DIM + (size_t)lane * 4;
        __builtin_prefetch(p_in + pb, 0, 0);
        __builtin_prefetch(m_in + pb, 0, 0);
        __builtin_prefetch(v_in + pb, 0, 0);
      }
      WAIT_ASYNC(3);
    } else {
      WAIT_ASYNC(0);
    }

    int slot = __builtin_amdgcn_readfirstlane(map[row]);  // wave-uniform -> scalar branch

    v4f p = lp[buf][w][lane];  // ds_load_b128
    v4f mm = lm[buf][w][lane];
    v4f vv = lv[buf][w][lane];

    if (slot >= 0) {
      const size_t gb = (size_t)slot * ADAM_DIM + (size_t)lane * 4;
      v4f g = __builtin_nontemporal_load((const v4f*)(g_in + gb));
      adam_update4(p, mm, vv, g, lr_t);
    }

    const size_t base = (size_t)row * ADAM_DIM + (size_t)lane * 4;
    __builtin_nontemporal_store(p, (v4f*)(p_out + base));
    __builtin_nontemporal_store(mm, (v4f*)(m_out + base));
    __builtin_nontemporal_store(vv, (v4f*)(v_out + base));
  }
}
#else
// Fallback: NT b128 loads + gfx1250 prefetch.
__global__ __launch_bounds__(256) void adam_rows_kernel(
    const float* __restrict__ p_in, const float* __restrict__ m_in,
    const float* __restrict__ v_in, const float* __restrict__ g_in,
    const int* __restrict__ map,
    float* __restrict__ p_out, float* __restrict__ m_out, float* __restrict__ v_out,
    int n_rows, float lr_t) {
  const int lane = threadIdx.x & 31;
  const int wave0 = blockIdx.x * WPB + (threadIdx.x >> 5);
  const int stride = gridDim.x * WPB;

  for (int row = wave0; row < n_rows; row += stride) {
    const size_t base = (size_t)row * ADAM_DIM + (size_t)lane * 4;
    const int nrow = row + stride;
    if (nrow < n_rows) {
      const size_t nb = (size_t)nrow * ADAM_DIM + (size_t)lane * 4;
      __builtin_prefetch(p_in + nb, 0, 0);
      __builtin_prefetch(m_in + nb, 0, 0);
      __builtin_prefetch(v_in + nb, 0, 0);
    }
    v4f p = __builtin_nontemporal_load((const v4f*)(p_in + base));
    v4f mm = __builtin_nontemporal_load((const v4f*)(m_in + base));
    v4f vv = __builtin_nontemporal_load((const v4f*)(v_in + base));
    int slot = __builtin_amdgcn_readfirstlane(map[row]);
    if (slot >= 0) {
      const size_t gb = (size_t)slot * ADAM_DIM + (size_t)lane * 4;
      v4f g = __builtin_nontemporal_load((const v4f*)(g_in + gb));
      adam_update4(p, mm, vv, g, lr_t);
    }
    __builtin_nontemporal_store(p, (v4f*)(p_out + base));
    __builtin_nontemporal_store(mm, (v4f*)(m_out + base));
    __builtin_nontemporal_store(vv, (v4f*)(v_out + base));
  }
}
#endif

// ------------------------------------------- fallback (if ws too small): copy + scatter
__global__ __launch_bounds__(256) void copy3_kernel(
    const float* __restrict__ a, const float* __restrict__ b, const float* __restrict__ c,
    float* __restrict__ oa, float* __restrict__ ob, float* __restrict__ oc, size_t n_vec) {
  size_t i = (size_t)blockIdx.x * 256 + threadIdx.x;
  const size_t stride = (size_t)gridDim.x * 256;
  for (; i < n_vec; i += stride) {
    __builtin_nontemporal_store(__builtin_nontemporal_load((const v4f*)a + i), (v4f*)oa + i);
    __builtin_nontemporal_store(__builtin_nontemporal_load((const v4f*)b + i), (v4f*)ob + i);
    __builtin_nontemporal_store(__builtin_nontemporal_load((const v4f*)c + i), (v4f*)oc + i);
  }
}

__global__ __launch_bounds__(256) void scatter_update_kernel(
    const float* __restrict__ p_in, const float* __restrict__ m_in,
    const float* __restrict__ v_in, const float* __restrict__ g_in,
    const long long* __restrict__ idx,
    float* __restrict__ p_out, float* __restrict__ m_out, float* __restrict__ v_out,
    int nnz, float lr_t) {
  const int lane = threadIdx.x & 31;
  const int j = blockIdx.x * WPB + (threadIdx.x >> 5);  // grad slot
  if (j >= nnz) return;
  int row = (int)idx[j];
  row = __builtin_amdgcn_readfirstlane(row);
  const size_t base = (size_t)row * ADAM_DIM + (size_t)lane * 4;
  const size_t gb = (size_t)j * ADAM_DIM + (size_t)lane * 4;

  v4f p = __builtin_nontemporal_load((const v4f*)(p_in + base));
  v4f mm = __builtin_nontemporal_load((const v4f*)(m_in + base));
  v4f vv = __builtin_nontemporal_load((const v4f*)(v_in + base));
  v4f g = __builtin_nontemporal_load((const v4f*)(g_in + gb));
  adam_update4(p, mm, vv, g, lr_t);
  __builtin_nontemporal_store(p, (v4f*)(p_out + base));
  __builtin_nontemporal_store(mm, (v4f*)(m_out + base));
  __builtin_nontemporal_store(vv, (v4f*)(v_out + base));
}

// ---------------------------------------------------------------- launcher
extern "C" void kernel_launch(void* const* d_in, const int* in_sizes, int n_in,
                              void* d_out, int out_size, void* d_ws, size_t ws_size,
                              hipStream_t stream) {
  const float* p = (const float*)d_in[0];
  const float* m = (const float*)d_in[1];
  const float* v = (const float*)d_in[2];
  const float* g = (const float*)d_in[3];
  const long long* idx = (const long long*)d_in[4];

  const int n_rows = in_sizes[0] / ADAM_DIM;
  const int nnz = in_sizes[4];

  float* po = (float*)d_out;
  float* mo = po + (size_t)n_rows * ADAM_DIM;
  float* vo = mo + (size_t)n_rows * ADAM_DIM;

  // Iteration-1 constants, f32 math to mirror JAX weak-typed promotion.
  const float bc1 = (float)(1.0 - 0.9);    // 0.1f
  const float bc2 = (float)(1.0 - 0.999);  // 0.001f
  const float lr_t = 0.001f * sqrtf(bc2) / bc1;

  if (ws_size >= (size_t)n_rows * sizeof(int)) {
    int* map = (int*)d_ws;
    init_map_kernel<<<(n_rows + 255) / 256, 256, 0, stream>>>(map, n_rows);
    build_map_kernel<<<(nnz + 255) / 256, 256, 0, stream>>>(idx, map, nnz);

    int blocks = (n_rows + WPB - 1) / WPB;
    if (blocks > 32768) blocks = 32768;  // grid-stride keeps the double-buffer pipeline live
    adam_rows_kernel<<<blocks, 256, 0, stream>>>(p, m, v, g, map, po, mo, vo, n_rows, lr_t);
  } else {
    // Fallback: full copy, then scatter-overwrite updated rows (indices unique).
    const size_t n_vec = (size_t)n_rows * ADAM_DIM / 4;
    size_t cblocks = (n_vec + 255) / 256;
    if (cblocks > 32768) cblocks = 32768;
    copy3_kernel<<<(unsigned)cblocks, 256, 0, stream>>>(p, m, v, po, mo, vo, n_vec);
    scatter_update_kernel<<<(nnz + WPB - 1) / WPB, 256, 0, stream>>>(
        p, m, v, g, idx, po, mo, vo, nnz, lr_t);
  }
}